// RCLoss_42554535969044
// MI455X (gfx1250) — compile-verified
//
#include <hip/hip_runtime.h>

typedef float v2f __attribute__((ext_vector_type(2)));
typedef float v8f __attribute__((ext_vector_type(8)));

#define IMH 96
#define IMW 96
#define NCH 256
#define NPATCH 94          // (96-3)/1+1 patches per row/col
#define CHUNK 64           // channels staged in LDS per pass
#define NCHUNK (NCH / CHUNK)
#define CSTRIDE 68         // floats between pixels (64 ch + 4 pad) -> 4-bank shift/pixel
#define RSTRIDE 6544       // floats between image rows (96*68 + 16) -> 16-bank shift/row
#define IN_LDS_FLOATS (3 * RSTRIDE)
#define NTHREADS 512
#define NWAVES 16
#define PPW 6              // ceil(94/16) patches per wave
#define SCR_PER_WAVE 96
#define ZPAD_OFF (2 * IN_LDS_FLOATS + NWAVES * SCR_PER_WAVE)  // 4-float zero pad
#define LDS_FLOATS (ZPAD_OFF + 4)
#define LDS_BYTES (LDS_FLOATS * 4)

__global__ __launch_bounds__(NTHREADS)
void rcloss_main(const float* __restrict__ pred, const float* __restrict__ targ,
                 float* __restrict__ wsum) {
  extern __shared__ float lds[];
  float* ldsP = lds;
  float* scr  = lds + 2 * IN_LDS_FLOATS;

  const int tid  = threadIdx.x;
  const int lane = tid & 31;
  const int wv   = tid >> 5;
  const int ph   = blockIdx.x;   // patch row 0..93
  const int b    = blockIdx.y;   // batch 0..3

  const size_t imgBase = (size_t)b * NCH * IMH * IMW;

  // Zero pad for the padding lanes (m >= 9): zeroed once, read-only afterwards.
  if (tid < 4) lds[ZPAD_OFF + tid] = 0.f;

  // Per-lane constants for the WMMA operand layout (A == B for a Gram):
  // lane holds pixel m = lane&15, channel pair klo = (lane<16 ? 0,1 : 2,3).
  const int m      = lane & 15;
  const int klo    = (lane < 16) ? 0 : 2;
  const bool mval  = (m < 9);
  const int pixOff = (m / 3) * RSTRIDE + (m % 3) * CSTRIDE + klo;
  const int step   = mval ? 4 : 0;   // floats advanced per K=4 chunk (0 for pad lanes)

  v8f accP[PPW];
  v8f accT[PPW];
  const v8f vzero = {0.f, 0.f, 0.f, 0.f, 0.f, 0.f, 0.f, 0.f};
#pragma unroll
  for (int j = 0; j < PPW; ++j) { accP[j] = vzero; accT[j] = vzero; }

  for (int ck = 0; ck < NCHUNK; ++ck) {
    __syncthreads();
    const int cbase = ck * CHUNK;
    // Cooperative stage: 64 channels x 3 rows x 96 cols for both inputs.
    for (int idx = tid; idx < CHUNK * 3 * IMW; idx += NTHREADS) {
      int ch  = idx / (3 * IMW);
      int pix = idx - ch * (3 * IMW);
      int row = pix / IMW;
      int col = pix - row * IMW;
      size_t g = imgBase + ((size_t)(cbase + ch) * IMH + (ph + row)) * IMW + col;
      int la = row * RSTRIDE + col * CSTRIDE + ch;
      ldsP[la]                 = pred[g];
      ldsP[la + IN_LDS_FLOATS] = targ[g];
    }
    __syncthreads();

#pragma unroll
    for (int j = 0; j < PPW; ++j) {
      const int pw = wv + NWAVES * j;
      if (pw < NPATCH) {
        // Pad lanes read a constant zeroed address; valid lanes walk channels.
        int ip = mval ? (pixOff + pw * CSTRIDE)                 : ZPAD_OFF;
        int it = mval ? (pixOff + pw * CSTRIDE + IN_LDS_FLOATS) : ZPAD_OFF;
#pragma unroll 4
        for (int c0 = 0; c0 < CHUNK; c0 += 4) {
          v2f a = *(const v2f*)(lds + ip);
          v2f t = *(const v2f*)(lds + it);
          ip += step;
          it += step;
          // D = A x B + C, A==B gives the uncentered Gram (fp32 WMMA).
          accP[j] = __builtin_amdgcn_wmma_f32_16x16x4_f32(
              false, a, false, a, (short)0, accP[j], false, false);
          accT[j] = __builtin_amdgcn_wmma_f32_16x16x4_f32(
              false, t, false, t, (short)0, accT[j], false, false);
        }
      }
    }
  }

  // Epilogue: Delta = Gp - Gt, centering correction, squared sum.
  float lsum = 0.f;
  float* sc = scr + wv * SCR_PER_WAVE;  // 81 entries + 9 row sums
  float* rb = sc + 81;
  const int N     = lane & 15;
  const int Mbase = (lane < 16) ? 0 : 8;

#pragma unroll
  for (int j = 0; j < PPW; ++j) {
    const int pw = wv + NWAVES * j;
    if (pw >= NPATCH) continue;
#pragma unroll
    for (int i = 0; i < 8; ++i) {
      int M = Mbase + i;
      if (M < 9 && N < 9) sc[M * 9 + N] = accP[j][i] - accT[j][i];
    }
    if (lane < 9) {
      float r = 0.f;
#pragma unroll
      for (int n = 0; n < 9; ++n) r += sc[lane * 9 + n];
      rb[lane] = r;
    }
    float S = 0.f;
#pragma unroll
    for (int n = 0; n < 9; ++n) S += rb[n];  // broadcast reads
    for (int e = lane; e < 81; e += 32) {
      int k = e / 9;
      int l = e - 9 * k;
      float v = sc[e] - (rb[k] + rb[l]) * (1.f / 9.f) + S * (1.f / 81.f);
      lsum += v * v;
    }
  }

  // wave32 reduction, then one atomic per wave.
  for (int off = 16; off > 0; off >>= 1)
    lsum += __shfl_xor(lsum, off, 32);
  if (lane == 0) atomicAdd(wsum, lsum);
}

__global__ void rcloss_finalize(const float* __restrict__ wsum,
                                float* __restrict__ out) {
  // mean over 4 * 94*94 * 81 = 2,862,864 elements
  out[0] = wsum[0] * (1.0f / 2862864.0f);
}

extern "C" void kernel_launch(void* const* d_in, const int* in_sizes, int n_in,
                              void* d_out, int out_size, void* d_ws, size_t ws_size,
                              hipStream_t stream) {
  const float* pred = (const float*)d_in[0];
  const float* targ = (const float*)d_in[1];
  float* ws = (float*)d_ws;

  (void)hipFuncSetAttribute((const void*)rcloss_main,
                            hipFuncAttributeMaxDynamicSharedMemorySize,
                            LDS_BYTES);
  (void)hipMemsetAsync(d_ws, 0, sizeof(float), stream);

  dim3 grid(NPATCH, 4);
  rcloss_main<<<grid, NTHREADS, LDS_BYTES, stream>>>(pred, targ, ws);
  rcloss_finalize<<<1, 1, 0, stream>>>(ws, (float*)d_out);
}